// Set2SetPooling_21251498180615
// MI455X (gfx1250) — compile-verified
//
#include <hip/hip_runtime.h>

typedef __attribute__((ext_vector_type(2))) float v2f;
typedef __attribute__((ext_vector_type(4))) float v4f;
typedef __attribute__((ext_vector_type(8))) float v8f;

// ---------------------------------------------------------------------------
// zero-init workspace state (hs, cs, q_star)
// ---------------------------------------------------------------------------
__global__ void init_zero(float* __restrict__ p, int n) {
  int i = blockIdx.x * blockDim.x + threadIdx.x;
  if (i < n) p[i] = 0.0f;
}

// ---------------------------------------------------------------------------
// batch is sorted; starts[g] = first node of graph g, starts[G] = N
// ---------------------------------------------------------------------------
__global__ void find_starts(const int* __restrict__ batch, int* __restrict__ starts,
                            int N, int G) {
  int i = blockIdx.x * blockDim.x + threadIdx.x;
  if (i >= N) return;
  int b = batch[i];
  int bp = (i == 0) ? -1 : batch[i - 1];
  for (int g = bp + 1; g <= b; ++g) starts[g] = i;
  if (i == N - 1) {
    for (int g = b + 1; g <= G; ++g) starts[g] = N;
  }
}

// ---------------------------------------------------------------------------
// gates[512,1024] = q_star[512,512] @ W_ih^T + hs[512,256] @ W_hh^T + (b_ih+b_hh)
// f32 WMMA 16x16x4, one wave per 16x16 output tile.
//   A (16x4): lane%16 = M row, K = (lane/16)*2 + v      (v = vgpr 0/1)
//   B (4x16) = W^T: lane%16 = N col, K = (lane/16)*2+v  (reads W row-major [N,K])
//   C/D (16x16): M = v + (lane/16)*8, N = lane%16
// ---------------------------------------------------------------------------
__global__ void __launch_bounds__(256)
gates_gemm(const float* __restrict__ qstar, const float* __restrict__ hs,
           const float* __restrict__ Wih, const float* __restrict__ Whh,
           const float* __restrict__ bih, const float* __restrict__ bhh,
           float* __restrict__ gates) {
  const int lane = threadIdx.x & 31;
  const int wave = blockIdx.x * (blockDim.x >> 5) + (threadIdx.x >> 5);
  const int TN = 1024 / 16;       // 64 tiles along N
  const int tm = wave / TN;       // 0..31  (512 rows)
  const int tn = wave % TN;       // 0..63
  const int r  = lane & 15;
  const int kh = (lane >> 4) << 1;  // 0 or 2

  const float* arow = qstar + (size_t)(tm * 16 + r) * 512;
  const float* brow = Wih   + (size_t)(tn * 16 + r) * 512;
  v8f acc = {};
  for (int k = 0; k < 512; k += 4) {
    v2f a, b;
    a.x = arow[k + kh]; a.y = arow[k + kh + 1];
    b.x = brow[k + kh]; b.y = brow[k + kh + 1];
    acc = __builtin_amdgcn_wmma_f32_16x16x4_f32(false, a, false, b, (short)0, acc,
                                                false, false);
  }
  const float* arow2 = hs  + (size_t)(tm * 16 + r) * 256;
  const float* brow2 = Whh + (size_t)(tn * 16 + r) * 256;
  for (int k = 0; k < 256; k += 4) {
    v2f a, b;
    a.x = arow2[k + kh]; a.y = arow2[k + kh + 1];
    b.x = brow2[k + kh]; b.y = brow2[k + kh + 1];
    acc = __builtin_amdgcn_wmma_f32_16x16x4_f32(false, a, false, b, (short)0, acc,
                                                false, false);
  }
  const int mb = tm * 16 + (lane >> 4) * 8;
  const int n  = tn * 16 + r;
  const float bias = bih[n] + bhh[n];
#pragma unroll
  for (int v = 0; v < 8; ++v)
    gates[(size_t)(mb + v) * 1024 + n] = acc[v] + bias;
}

// ---------------------------------------------------------------------------
// LSTM state update (PyTorch gate order i,f,g,o); also writes q-half of q_star
// ---------------------------------------------------------------------------
__device__ __forceinline__ float sigmoidf_(float x) {
  return 1.0f / (1.0f + __expf(-x));
}

__global__ void __launch_bounds__(256)
lstm_update(const float* __restrict__ gates, float* __restrict__ hs,
            float* __restrict__ cs, float* __restrict__ qstar) {
  int idx = blockIdx.x * blockDim.x + threadIdx.x;   // exactly 512*256 threads
  int b = idx >> 8, c = idx & 255;
  const float* g = gates + (size_t)b * 1024;
  float ig = sigmoidf_(g[c]);
  float fg = sigmoidf_(g[256 + c]);
  float gg = tanhf(g[512 + c]);
  float og = sigmoidf_(g[768 + c]);
  float cn = fg * cs[idx] + ig * gg;
  cs[idx] = cn;
  float h = og * tanhf(cn);
  hs[idx] = h;
  qstar[(size_t)b * 512 + c] = h;
}

// ---------------------------------------------------------------------------
// Per-graph attention: single-pass online softmax + weighted readout.
// One block (8 waves) per graph; wave w handles nodes start+w, start+w+8, ...
// Lane l owns columns [4l..4l+3] and [128+4l..128+4l+3]  -> two b128 loads
// per node per lane (4x fewer vmem instructions than strided b32).
// Writes r-half of q_star.
// ---------------------------------------------------------------------------
__global__ void __launch_bounds__(256)
attention(const float* __restrict__ x, const int* __restrict__ starts,
          const float* __restrict__ hs, float* __restrict__ qstar) {
  const int g    = blockIdx.x;
  const int tid  = threadIdx.x;
  const int lane = tid & 31;
  const int wv   = tid >> 5;          // 0..7

  __shared__ __align__(16) float q[256];
  __shared__ __align__(16) float rall[8][256];
  __shared__ float wmax[8];
  __shared__ float wsum[8];

  q[tid] = hs[(size_t)g * 256 + tid];
  __syncthreads();

  const int start = starts[g];
  const int end   = starts[g + 1];

  const v4f q0 = *(const v4f*)&q[4 * lane];
  const v4f q1 = *(const v4f*)&q[128 + 4 * lane];

  float m = -__builtin_inff();
  float s = 0.0f;
  v4f r0 = {}, r1 = {};

  for (int i = start + wv; i < end; i += 8) {
    const v4f* xr = (const v4f*)(x + (size_t)i * 256);
    v4f x0 = xr[lane];        // global_load_b128
    v4f x1 = xr[32 + lane];   // global_load_b128
    float d = 0.0f;
    d = fmaf(x0.x, q0.x, d); d = fmaf(x0.y, q0.y, d);
    d = fmaf(x0.z, q0.z, d); d = fmaf(x0.w, q0.w, d);
    d = fmaf(x1.x, q1.x, d); d = fmaf(x1.y, q1.y, d);
    d = fmaf(x1.z, q1.z, d); d = fmaf(x1.w, q1.w, d);
#pragma unroll
    for (int off = 16; off; off >>= 1) d += __shfl_xor(d, off, 32);
    // d is now wave-uniform: full dot(x_i, q)
    if (d > m) {
      float sc = (m == -__builtin_inff()) ? 0.0f : __expf(m - d);
      s *= sc;
      r0.x *= sc; r0.y *= sc; r0.z *= sc; r0.w *= sc;
      r1.x *= sc; r1.y *= sc; r1.z *= sc; r1.w *= sc;
      m = d;
    }
    float a = __expf(d - m);
    s += a;
    r0.x = fmaf(a, x0.x, r0.x); r0.y = fmaf(a, x0.y, r0.y);
    r0.z = fmaf(a, x0.z, r0.z); r0.w = fmaf(a, x0.w, r0.w);
    r1.x = fmaf(a, x1.x, r1.x); r1.y = fmaf(a, x1.y, r1.y);
    r1.z = fmaf(a, x1.z, r1.z); r1.w = fmaf(a, x1.w, r1.w);
  }

  *(v4f*)&rall[wv][4 * lane]       = r0;   // ds_store_b128
  *(v4f*)&rall[wv][128 + 4 * lane] = r1;   // ds_store_b128
  if (lane == 0) { wmax[wv] = m; wsum[wv] = s; }
  __syncthreads();

  float M = -__builtin_inff();
#pragma unroll
  for (int w = 0; w < 8; ++w) M = fmaxf(M, wmax[w]);

  float stot = 0.0f, r = 0.0f;
#pragma unroll
  for (int w = 0; w < 8; ++w) {
    float sw = wsum[w];
    if (sw > 0.0f) {
      float sc = __expf(wmax[w] - M);
      stot = fmaf(sw, sc, stot);
      r    = fmaf(rall[w][tid], sc, r);
    }
  }
  qstar[(size_t)g * 512 + 256 + tid] = (stot > 0.0f) ? (r / stot) : 0.0f;
}

// ---------------------------------------------------------------------------
// out[512,256] = q_star[512,512] @ W_out^T + b_out   (f32 WMMA)
// ---------------------------------------------------------------------------
__global__ void __launch_bounds__(256)
out_gemm(const float* __restrict__ qstar, const float* __restrict__ Wout,
         const float* __restrict__ bout, float* __restrict__ out) {
  const int lane = threadIdx.x & 31;
  const int wave = blockIdx.x * (blockDim.x >> 5) + (threadIdx.x >> 5);
  const int TN = 256 / 16;        // 16 tiles along N
  const int tm = wave / TN;       // 0..31
  const int tn = wave % TN;
  const int r  = lane & 15;
  const int kh = (lane >> 4) << 1;

  const float* arow = qstar + (size_t)(tm * 16 + r) * 512;
  const float* brow = Wout  + (size_t)(tn * 16 + r) * 512;
  v8f acc = {};
  for (int k = 0; k < 512; k += 4) {
    v2f a, b;
    a.x = arow[k + kh]; a.y = arow[k + kh + 1];
    b.x = brow[k + kh]; b.y = brow[k + kh + 1];
    acc = __builtin_amdgcn_wmma_f32_16x16x4_f32(false, a, false, b, (short)0, acc,
                                                false, false);
  }
  const int mb = tm * 16 + (lane >> 4) * 8;
  const int n  = tn * 16 + r;
  const float bias = bout[n];
#pragma unroll
  for (int v = 0; v < 8; ++v)
    out[(size_t)(mb + v) * 256 + n] = acc[v] + bias;
}

// ---------------------------------------------------------------------------
extern "C" void kernel_launch(void* const* d_in, const int* in_sizes, int n_in,
                              void* d_out, int out_size, void* d_ws, size_t ws_size,
                              hipStream_t stream) {
  const float* x     = (const float*)d_in[0];   // [N,256]
  const int*   batch = (const int*)  d_in[1];   // [N] sorted
  const float* Wih   = (const float*)d_in[2];   // [1024,512]
  const float* Whh   = (const float*)d_in[3];   // [1024,256]
  const float* bih   = (const float*)d_in[4];   // [1024]
  const float* bhh   = (const float*)d_in[5];   // [1024]
  const float* Wout  = (const float*)d_in[6];   // [256,512]
  const float* bout  = (const float*)d_in[7];   // [256]
  const int N = in_sizes[0] / 256;
  const int G = 512;

  float* ws    = (float*)d_ws;
  float* hs    = ws;                          // 512*256
  float* cs    = hs + 512 * 256;              // 512*256
  float* qstar = cs + 512 * 256;              // 512*512
  float* gates = qstar + 512 * 512;           // 512*1024
  int*   starts = (int*)(gates + 512 * 1024); // G+1

  // zero hs, cs, qstar (contiguous)
  int zn = 512 * 256 * 2 + 512 * 512;
  init_zero<<<(zn + 255) / 256, 256, 0, stream>>>(ws, zn);
  find_starts<<<(N + 255) / 256, 256, 0, stream>>>(batch, starts, N, G);

  for (int step = 0; step < 3; ++step) {
    gates_gemm<<<256, 256, 0, stream>>>(qstar, hs, Wih, Whh, bih, bhh, gates);
    lstm_update<<<512, 256, 0, stream>>>(gates, hs, cs, qstar);
    attention<<<G, 256, 0, stream>>>(x, starts, hs, qstar);
  }
  out_gemm<<<64, 256, 0, stream>>>(qstar, Wout, bout, (float*)d_out);
}